// VDP_Flatten_26499948216838
// MI455X (gfx1250) — compile-verified
//
#include <hip/hip_runtime.h>

// VDP_Flatten: mu [4,32,16,16] -> mu_flat [4,8192] (BHWC flatten)
//              sigma [4,32,256,256] -> sigma_flat [4,8192,8192] block-diagonal
// d_out = mu_flat (32768 f32) ++ sigma_flat (268435456 f32)
//
// Pure data movement: ~1.074 GB of stores + 33.5 MB of loads => ~47 us floor
// at 23.3 TB/s. Store-bandwidth bound; optimized with b128 nontemporal stores
// (output >> 192 MB L2) and conflict-free LDS transpose for mu.

typedef float __attribute__((ext_vector_type(4))) floatx4;

#define BN   4
#define CN   32
#define HWN  256
#define ROW  8192   /* CN*HWN */
#define LDSP 257    /* padded LDS row stride (odd -> no bank conflicts) */

// One block (256 threads) per batch element: transpose [C=32, HW=256] ->
// [HW=256, C=32] through LDS so both global read and write are coalesced.
__global__ void mu_flatten_kernel(const float* __restrict__ mu,
                                  float* __restrict__ out) {
  __shared__ float lds[CN * LDSP];
  const int b   = blockIdx.x;
  const int tid = threadIdx.x;                 // 256 threads
  const float* __restrict__ src = mu  + b * (CN * HWN);
  float*       __restrict__ dst = out + b * (CN * HWN);

  // Coalesced global load: idx = c*256 + hw, consecutive lanes -> consecutive hw
#pragma unroll
  for (int k = 0; k < CN; ++k) {
    int idx = tid + (k << 8);
    int c   = idx >> 8;
    int hw  = idx & (HWN - 1);
    lds[c * LDSP + hw] = src[idx];
  }
  __syncthreads();

  // Coalesced global store: o = hw*32 + c; LDS read stride 257 -> conflict-free
#pragma unroll
  for (int k = 0; k < CN; ++k) {
    int o  = tid + (k << 8);
    int c  = o & (CN - 1);
    int hw = o >> 5;
    __builtin_nontemporal_store(lds[c * LDSP + hw], dst + o);
  }
}

// One block (256 threads) per output row of sigma_flat.
// gridDim.x = B*8192. Row r = c*256 + i holds sigma[b,c,i,:] in columns
// [c*256, c*256+256) and zeros elsewhere. Each thread writes 8 float4 (b128)
// nontemporal stores; the data-vs-zero branch is uniform across a wave32.
__global__ void sigma_blockdiag_kernel(const float* __restrict__ sigma,
                                       float* __restrict__ out) {
  const int gb = blockIdx.x;           // 0 .. B*ROW-1
  const int b  = gb >> 13;             // / 8192
  const int r  = gb & (ROW - 1);
  const int c  = r >> 8;               // channel block of this row
  const int i  = r & (HWN - 1);

  // Source row sigma[b,c,i,0..255] viewed as 64 float4 (16B-aligned).
  const floatx4* __restrict__ src4 =
      (const floatx4*)(sigma + (size_t)(((b * CN + c) * HWN + i)) * HWN);
  // Destination row of sigma_flat (float offset is multiple of 8192 -> aligned).
  floatx4* __restrict__ dst4 = (floatx4*)(out + (size_t)gb * ROW);

  const floatx4 zero = {0.0f, 0.0f, 0.0f, 0.0f};
  const int tid = threadIdx.x;

#pragma unroll
  for (int k = 0; k < 8; ++k) {
    int f = tid + (k << 8);            // float4 column index, 0..2047
    floatx4 v = zero;
    if ((f >> 6) == c) {               // inside the diagonal block (wave-uniform)
      v = __builtin_nontemporal_load(src4 + (f & 63));
    }
    __builtin_nontemporal_store(v, dst4 + f);
  }
}

extern "C" void kernel_launch(void* const* d_in, const int* in_sizes, int n_in,
                              void* d_out, int out_size, void* d_ws, size_t ws_size,
                              hipStream_t stream) {
  const float* mu    = (const float*)d_in[0];   // [4,32,16,16]
  const float* sigma = (const float*)d_in[1];   // [4,32,256,256]
  float* out = (float*)d_out;

  float* mu_flat    = out;                       // first 4*8192 floats
  float* sigma_flat = out + BN * CN * HWN;       // remaining 4*8192*8192 floats

  mu_flatten_kernel<<<BN, 256, 0, stream>>>(mu, mu_flat);
  sigma_blockdiag_kernel<<<BN * ROW, 256, 0, stream>>>(sigma, sigma_flat);
}